// GraphSAGEModelPlaceholder_13340168421671
// MI455X (gfx1250) — compile-verified
//
#include <hip/hip_runtime.h>

typedef __attribute__((ext_vector_type(2))) float v2f;
typedef __attribute__((ext_vector_type(4))) float f4;
typedef __attribute__((ext_vector_type(8))) float v8f;

#define BLOCK_M  32
#define D_FEAT   256
#define UNITS    64
#define NTHREADS 256

// Fused GraphSAGE layer:
//   combined = [ h (D_H) | mean_k feat[nbr] (256) ]   (built in LDS)
//   out      = relu(combined @ W + b)                 (WMMA f32 16x16x4)
template <int D_H, int K_NBR, bool GATHER_H>
__global__ __launch_bounds__(NTHREADS)
void sage_layer_kernel(const float* __restrict__ feat,
                       const int*   __restrict__ target_ids,
                       const float* __restrict__ h_in,
                       const int*   __restrict__ nbr_ids,
                       const float* __restrict__ W,      // [D_H+256, 64] row-major
                       const float* __restrict__ bias,   // [64]
                       float*       __restrict__ out,    // [B, 64]
                       int n_nodes)
{
    constexpr int KTOT   = D_H + D_FEAT;   // GEMM K dimension
    constexpr int STRIDE = KTOT + 4;       // +4 floats: stride % 64 banks == 4 -> conflict-free
    extern __shared__ float lds[];         // [BLOCK_M][STRIDE]

    const int tid  = threadIdx.x;
    const int row0 = blockIdx.x * BLOCK_M;

    // ---------- Stage 1a: self part of "combined" ----------
    if constexpr (GATHER_H) {
        for (int idx = tid; idx < BLOCK_M * (D_H / 4); idx += NTHREADS) {
            const int r  = idx / (D_H / 4);
            const int c4 = idx % (D_H / 4);
            const int node = target_ids[row0 + r];
            f4 v = *(reinterpret_cast<const f4*>(feat + (size_t)node * D_FEAT) + c4);
            *reinterpret_cast<f4*>(&lds[r * STRIDE + c4 * 4]) = v;
        }
    } else {
        for (int idx = tid; idx < BLOCK_M * (D_H / 4); idx += NTHREADS) {
            const int r  = idx / (D_H / 4);
            const int c4 = idx % (D_H / 4);
            f4 v = *(reinterpret_cast<const f4*>(h_in + (size_t)(row0 + r) * D_H) + c4);
            *reinterpret_cast<f4*>(&lds[r * STRIDE + c4 * 4]) = v;
        }
    }

    // ---------- Stage 1b: neighbor mean (fixed denominator K) ----------
    constexpr float invK = 1.0f / (float)K_NBR;
    for (int idx = tid; idx < BLOCK_M * (D_FEAT / 4); idx += NTHREADS) {
        const int r  = idx / (D_FEAT / 4);
        const int c4 = idx % (D_FEAT / 4);
        f4 acc = {0.f, 0.f, 0.f, 0.f};
#pragma unroll
        for (int k = 0; k < K_NBR; ++k) {
            const int node = nbr_ids[(size_t)(row0 + r) * K_NBR + k];
            if ((unsigned)node < (unsigned)n_nodes) {
                acc += *(reinterpret_cast<const f4*>(feat + (size_t)node * D_FEAT) + c4);
            }
        }
        acc *= invK;
        *reinterpret_cast<f4*>(&lds[r * STRIDE + D_H + c4 * 4]) = acc;
    }

    __syncthreads();

    // ---------- Stage 2: WMMA GEMM  [BLOCK_M x KTOT] @ [KTOT x 64] ----------
    // 8 tiles of 16x16 (2 M-tiles x 4 N-tiles), one per wave. Uniform control flow.
    const int wave  = tid >> 5;
    const int lane  = tid & 31;
    const int lrow  = lane & 15;          // M (A/C) or N (B) index within tile
    const int khalf = (lane >> 4) << 1;   // lanes 0-15 -> K+0/K+1, lanes 16-31 -> K+2/K+3
    const int mt    = wave >> 2;
    const int nt    = wave & 3;

    const float* Arow = &lds[(mt * 16 + lrow) * STRIDE];
    const float* Bcol = W + (nt * 16 + lrow);

    v8f c = {};
    for (int k0 = 0; k0 < KTOT; k0 += 4) {
        // A 16x4 f32 fragment: lane holds row M=lrow, (K=khalf, khalf+1) -> ds_load_b64
        v2f a = *reinterpret_cast<const v2f*>(Arow + k0 + khalf);
        // B 4x16 f32 fragment: VGPR0 = rows K=0/K=2, VGPR1 = rows K=1/K=3
        v2f b;
        b.x = Bcol[(size_t)(k0 + khalf)     * UNITS];
        b.y = Bcol[(size_t)(k0 + khalf + 1) * UNITS];
        c = __builtin_amdgcn_wmma_f32_16x16x4_f32(
                /*neg_a=*/false, a, /*neg_b=*/false, b,
                /*c_mod=*/(short)0, c, /*reuse_a=*/false, /*reuse_b=*/false);
    }

    // ---------- bias + relu + store (C/D layout: VGPR j -> M=j / M=j+8) ----------
    const int   n     = nt * 16 + lrow;
    const float bv    = bias[n];
    const int   mbase = mt * 16 + ((lane >> 4) << 3);
#pragma unroll
    for (int j = 0; j < 8; ++j) {
        float v = c[j] + bv;
        v = v > 0.f ? v : 0.f;
        out[(size_t)(row0 + mbase + j) * UNITS + n] = v;
    }
}

extern "C" void kernel_launch(void* const* d_in, const int* in_sizes, int n_in,
                              void* d_out, int out_size, void* d_ws, size_t ws_size,
                              hipStream_t stream)
{
    const int*   target = (const int*)  d_in[0];   // [B]
    const int*   nbr1   = (const int*)  d_in[1];   // [B,10]
    const int*   nbr2   = (const int*)  d_in[2];   // [B,5]
    const float* feat   = (const float*)d_in[3];   // [N,256]
    const float* W1     = (const float*)d_in[4];   // [512,64]
    const float* b1     = (const float*)d_in[5];   // [64]
    const float* W2     = (const float*)d_in[6];   // [320,64]
    const float* b2     = (const float*)d_in[7];   // [64]

    float* out = (float*)d_out;                    // [B,64]
    float* h1  = (float*)d_ws;                     // [B,64] intermediate (4 MB)

    const int batch   = in_sizes[0];
    const int n_nodes = in_sizes[3] / D_FEAT;

    dim3 grid(batch / BLOCK_M), block(NTHREADS);

    const size_t lds1 = (size_t)BLOCK_M * (256 + D_FEAT + 4) * sizeof(float); // 66 KB
    hipLaunchKernelGGL((sage_layer_kernel<256, 10, true>), grid, block, lds1, stream,
                       feat, target, (const float*)nullptr, nbr1, W1, b1, h1, n_nodes);

    const size_t lds2 = (size_t)BLOCK_M * (64 + D_FEAT + 4) * sizeof(float);  // 41.5 KB
    hipLaunchKernelGGL((sage_layer_kernel<64, 5, false>), grid, block, lds2, stream,
                       feat, (const int*)nullptr, h1, nbr2, W2, b2, out, n_nodes);
}